// CausalBiasingNetwork_71906342469953
// MI455X (gfx1250) — compile-verified
//
#include <hip/hip_runtime.h>
#include <hip/hip_bf16.h>

// CausalBiasingNetwork for MI455X (gfx1250, wave32).
// bias[b,s,t] = sum_r scaled[b,r,s]*u[b,r,t] + v[b,t]
// Entirely memory-bound by the 268MB fp32 output; all matrix math uses
// V_WMMA_F32_16X16X4_F32 (fp32 in/out, matches reference semantics).

#define BB 4
#define SS 4096
#define HH 1024
#define RR 16

typedef float v2f __attribute__((ext_vector_type(2)));
typedef float v8f __attribute__((ext_vector_type(8)));

#define WMMA_F32_K4(a, b, c) \
    __builtin_amdgcn_wmma_f32_16x16x4_f32(false, (a), false, (b), (short)0, (c), false, false)

// ---------------------------------------------------------------------------
// Kernel 0: be_sum[h] = sum_r be[r][h]
// ---------------------------------------------------------------------------
__global__ void cbn_besum_kernel(const float* __restrict__ be, float* __restrict__ be_sum) {
    int h = blockIdx.x * blockDim.x + threadIdx.x;
    if (h < HH) {
        float s = 0.f;
#pragma unroll
        for (int r = 0; r < RR; ++r) s += be[r * HH + h];
        be_sum[h] = s;
    }
}

// ---------------------------------------------------------------------------
// Kernel 1: per 16-row tile of hs, compute via WMMA f32 16x16x4:
//   scaledT[b][s][r] = (hs[b,s,:]·wc[r] + bc[r]) * strength[r]
//   uT[b][t][r]      =  hs[b,t,:]·we[r]
//   v[b][s]          =  hs[b,s,:]·be_sum        (lane-pair reduction)
// One wave per 16-row tile; A = hs tile (M=s, K=h), B = wc^T / we^T (K=h, N=r).
// ---------------------------------------------------------------------------
__global__ void cbn_phase1_kernel(const float* __restrict__ hs,
                                  const float* __restrict__ wc,
                                  const float* __restrict__ bc,
                                  const float* __restrict__ we,
                                  const float* __restrict__ strength,
                                  const float* __restrict__ be_sum,
                                  float* __restrict__ scaledT,
                                  float* __restrict__ uT,
                                  float* __restrict__ vv) {
    const int wave = threadIdx.x >> 5;
    const int lane = threadIdx.x & 31;
    const int half = lane >> 4;   // 0: K even pair, 1: K odd pair (per ISA A/B layout)
    const int m    = lane & 15;   // A row (s) == B col (r) lane index

    const int tile = blockIdx.x * 8 + wave;          // 0 .. B*S/16-1 (=1023)
    const int b    = tile >> 8;                      // / (S/16)
    const int s0   = (tile & 255) << 4;

    const float* hsrow = hs + ((size_t)b * SS + s0 + m) * HH;  // A: row m of tile
    const float* wcrow = wc + (size_t)m * HH;                  // B: column r=m
    const float* werow = we + (size_t)m * HH;

    v8f cs = {};   // scores tile  (M=s, N=r)
    v8f cu = {};   // u tile       (M=t, N=r)
    float vacc = 0.f;

    for (int k0 = 0; k0 < HH; k0 += 4) {
        const int ko = k0 + 2 * half;
        v2f a  = *(const v2f*)(hsrow + ko);
        v2f bw = *(const v2f*)(wcrow + ko);
        v2f bu = *(const v2f*)(werow + ko);
        v2f bs = *(const v2f*)(be_sum + ko);
        cs = WMMA_F32_K4(a, bw, cs);
        cu = WMMA_F32_K4(a, bu, cu);
        vacc += a.x * bs.x + a.y * bs.y;   // partial dot for row m (this K half)
    }

    const float bcn  = bc[m];
    const float strn = strength[m];

    // C layout: element e of lane -> row (e + 8*half) of tile, col N = m.
    float* srow = scaledT + ((size_t)b * SS + s0 + 8 * half) * RR + m;
    float* urow = uT      + ((size_t)b * SS + s0 + 8 * half) * RR + m;
#pragma unroll
    for (int e = 0; e < 8; ++e) {
        srow[e * RR] = (cs[e] + bcn) * strn;
        urow[e * RR] = cu[e];
    }

    // v: combine the two K-halves held by lanes L and L+16.
    float vtot = vacc + __shfl_xor(vacc, 16, 32);
    if (half == 0) vv[(size_t)b * SS + s0 + m] = vtot;
}

// ---------------------------------------------------------------------------
// Kernel 2: bias[b, s0:s0+16, t0:t0+16] = A(16x16) x B(16x16) + v, K=16 via
// four chained V_WMMA_F32_16X16X4_F32. Each wave does a 16x64 strip (4 t-tiles)
// reusing its A registers; output stream (256MB) is the bottleneck.
// ---------------------------------------------------------------------------
__global__ void cbn_phase2_kernel(const float* __restrict__ scaledT,
                                  const float* __restrict__ uT,
                                  const float* __restrict__ vv,
                                  float* __restrict__ bias) {
    const int wave = threadIdx.x >> 5;
    const int lane = threadIdx.x & 31;
    const int half = lane >> 4;
    const int m    = lane & 15;

    const int job = blockIdx.x * 8 + wave;   // 0 .. B*256*64-1 (=65535)
    const int b   = job >> 14;               // / (256*64)
    const int rem = job & 16383;
    const int s0  = (rem >> 6) << 4;         // s-tile base
    const int t0b = (rem & 63) << 6;         // 64-wide t strip base

    // A: scaledT[b][s0+m][k], per-chunk K window 4j + 2*half + {0,1}
    const float* arow = scaledT + ((size_t)b * SS + s0 + m) * RR + 2 * half;
    v2f a0 = *(const v2f*)(arow + 0);
    v2f a1 = *(const v2f*)(arow + 4);
    v2f a2 = *(const v2f*)(arow + 8);
    v2f a3 = *(const v2f*)(arow + 12);

#pragma unroll
    for (int tt = 0; tt < 4; ++tt) {
        const int t0 = t0b + tt * 16;
        const float* brow = uT + ((size_t)b * SS + t0 + m) * RR + 2 * half;
        v2f b0 = *(const v2f*)(brow + 0);
        v2f b1 = *(const v2f*)(brow + 4);
        v2f b2 = *(const v2f*)(brow + 8);
        v2f b3 = *(const v2f*)(brow + 12);

        v8f c = {};
        c = WMMA_F32_K4(a0, b0, c);
        c = WMMA_F32_K4(a1, b1, c);
        c = WMMA_F32_K4(a2, b2, c);
        c = WMMA_F32_K4(a3, b3, c);

        const float vt = vv[(size_t)b * SS + t0 + m];   // depends only on t (N=m)

        float* out = bias + ((size_t)b * SS + s0 + 8 * half) * SS + t0 + m;
#pragma unroll
        for (int e = 0; e < 8; ++e) out[(size_t)e * SS] = c[e] + vt;
    }
}

// ---------------------------------------------------------------------------
extern "C" void kernel_launch(void* const* d_in, const int* in_sizes, int n_in,
                              void* d_out, int out_size, void* d_ws, size_t ws_size,
                              hipStream_t stream) {
    const float* hs       = (const float*)d_in[0];   // [B,S,H]
    const float* wc       = (const float*)d_in[1];   // [R,H]
    const float* bc       = (const float*)d_in[2];   // [R]
    const float* we       = (const float*)d_in[3];   // [R,H]
    const float* be       = (const float*)d_in[4];   // [R,H]
    const float* strength = (const float*)d_in[5];   // [R]
    float* bias = (float*)d_out;                     // [B,S,S]

    // Workspace layout (floats): scaledT[B*S*16] | uT[B*S*16] | v[B*S] | be_sum[H]
    float* ws      = (float*)d_ws;
    float* scaledT = ws;
    float* uT      = ws + (size_t)BB * SS * RR;
    float* vv      = ws + (size_t)2 * BB * SS * RR;
    float* be_sum  = vv + (size_t)BB * SS;

    // Kernel 0: be_sum (H=1024 threads)
    cbn_besum_kernel<<<HH / 256, 256, 0, stream>>>(be, be_sum);

    // Kernel 1: B*S/16 = 1024 wave-tiles, 8 waves/block -> 128 blocks
    cbn_phase1_kernel<<<(BB * SS / 16) / 8, 256, 0, stream>>>(
        hs, wc, bc, we, strength, be_sum, scaledT, uT, vv);

    // Kernel 2: B * 256 s-tiles * 64 t-strips = 65536 wave-jobs -> 8192 blocks
    cbn_phase2_kernel<<<(BB * 256 * 64) / 8, 256, 0, stream>>>(scaledT, uT, vv, bias);
}